// MoEsparseRouting_78632261255992
// MI455X (gfx1250) — compile-verified
//
#include <hip/hip_runtime.h>
#include <hip/hip_bf16.h>

// ---------------------------------------------------------------------------
// MoE with hard gumbel routing + tensor-train expert, CDNA5 (gfx1250).
// Forward value of gumbel_softmax(hard=True) is an exact one-hot => per batch
// we select one expert and run 6 chained small GEMMs, all mapped onto
// V_WMMA_F32_16X16X4_F32 (the fp32 WMMA shape on CDNA5, wave32).
// Expert cores are staged global->LDS via GLOBAL_LOAD_ASYNC_TO_LDS (ASYNCcnt)
// when the builtins are available, else through VGPR copies.
// ---------------------------------------------------------------------------

typedef float v2f __attribute__((ext_vector_type(2)));
typedef float v8f __attribute__((ext_vector_type(8)));
typedef int v4i __attribute__((__vector_size__(4 * sizeof(int))));

#define AS1 __attribute__((address_space(1)))
#define AS3 __attribute__((address_space(3)))

#if defined(__HIP_DEVICE_COMPILE__) && \
    __has_builtin(__builtin_amdgcn_global_load_async_to_lds_b32) && \
    __has_builtin(__builtin_amdgcn_global_load_async_to_lds_b128) && \
    __has_builtin(__builtin_amdgcn_s_wait_asynccnt)
#define USE_ASYNC_LDS 1
#else
#define USE_ASYNC_LDS 0
#endif

// D = A(16x4) * B(4x16) + C(16x16), fp32 WMMA.
__device__ __forceinline__ v8f wmma4(v2f a, v2f b, v8f c) {
  return __builtin_amdgcn_wmma_f32_16x16x4_f32(
      /*neg_a=*/false, a, /*neg_b=*/false, b,
      /*c_mod=*/(short)0, c, /*reuse_a=*/false, /*reuse_b=*/false);
}

// A fragment: A[m][k], row stride lda floats.  ISA layout (16x4 f32):
// lanes 0-15: M=lane, K={0,1}; lanes 16-31: M=lane-16, K={2,3}.
__device__ __forceinline__ v2f load_a(const float* p, int lda) {
  int lane = threadIdx.x & 31;
  int m = lane & 15, kh = (lane >> 4) << 1;
  v2f a;
  a.x = p[m * lda + kh];
  a.y = p[m * lda + kh + 1];
  return a;
}

// B fragment: B[k][n], row stride ldb floats.  Rows striped across lanes:
// VGPR0 = rows {0,2}, VGPR1 = rows {1,3} (lane halves), N = lane&15.
__device__ __forceinline__ v2f load_b(const float* p, int ldb) {
  int lane = threadIdx.x & 31;
  int n = lane & 15, kh = (lane >> 4) << 1;
  v2f b;
  b.x = p[kh * ldb + n];
  b.y = p[(kh + 1) * ldb + n];
  return b;
}

// C/D fragment: lane l holds D[8*(l>>4)+i][l&15], i=0..7.
__device__ __forceinline__ void store_c(float* p, int ldc, v8f c) {
  int lane = threadIdx.x & 31;
  int n = lane & 15, mb = (lane >> 4) << 3;
#pragma unroll
  for (int i = 0; i < 8; ++i) p[(mb + i) * ldc + n] = c[i];
}

__device__ __forceinline__ v8f zero8() {
  v8f z = {0.f, 0.f, 0.f, 0.f, 0.f, 0.f, 0.f, 0.f};
  return z;
}

// ---------------------------------------------------------------------------
// Router: pooled = mean_s X[b,s,:]; logits = pooled @ W^T + b + gumbel(u);
// expert_idx[b] = argmax (hard one-hot forward value).
// ---------------------------------------------------------------------------
__global__ void router_kernel(const float* __restrict__ X,
                              const float* __restrict__ rw,
                              const float* __restrict__ rb,
                              const float* __restrict__ un,
                              int* __restrict__ eidx) {
  __shared__ float pooled[768];
  __shared__ float slog[4];
  const int b = blockIdx.x;
  const int tid = threadIdx.x;

  for (int j = tid; j < 768; j += 256) {
    const float* xp = X + (size_t)b * 512 * 768 + j;
    float s = 0.f;
    for (int t = 0; t < 512; ++t) s += xp[(size_t)t * 768];
    pooled[j] = s * (1.0f / 512.0f);
  }
  __syncthreads();

  if (tid < 4) {
    float d = rb[tid];
    const float* wrow = rw + tid * 768;
    for (int j = 0; j < 768; ++j) d += pooled[j] * wrow[j];
    float u = un[b * 4 + tid];
    float g = -__logf(-__logf(u + 1e-10f) + 1e-10f);
    slog[tid] = d + g;  // tau=1: argmax unaffected by scaling
  }
  __syncthreads();

  if (tid == 0) {
    int best = 0;
    float bv = slog[0];
    for (int e = 1; e < 4; ++e)
      if (slog[e] > bv) { bv = slog[e]; best = e; }
    eidx[b] = best;
  }
}

// ---------------------------------------------------------------------------
// TT contraction. One workgroup = (batch, 16-token tile). 256 thr = 8 waves.
// Dynamic LDS layout (floats):
//   sC0[12*32] sC1[256*32] sC2[256*32] sC3[32*256] sC4[32*256] sC5[32*16]
//   sT0[1024*32] (reused as sT4[128*32])  sT1[16*256]  sT2[16*32]  sT3[16*256]
// Total 75136 floats = 293.5 KB  (<= 320 KB per-workgroup LDS on CDNA5).
// ---------------------------------------------------------------------------
#define NWAVE 8

__global__ void tt_moe_kernel(const float* __restrict__ X,
                              const float* __restrict__ c0,
                              const float* __restrict__ c1,
                              const float* __restrict__ c2,
                              const float* __restrict__ c3,
                              const float* __restrict__ c4,
                              const float* __restrict__ c5,
                              const int* __restrict__ eidx,
                              float* __restrict__ out) {
  extern __shared__ float sm[];
  float* sC0 = sm;              // 384
  float* sC1 = sC0 + 384;       // 8192  [(c*32+p)][q]
  float* sC2 = sC1 + 8192;      // 8192  [(a*32+q)][r]
  float* sC3 = sC2 + 8192;      // 8192  [r][n0*32+u]
  float* sC4 = sC3 + 8192;      // 8192  [u][n1*32+v]
  float* sC5 = sC4 + 8192;      // 512   [v][n2] padded to 16 cols
  float* sT0 = sC5 + 512;       // 32768 [s][a][c][p]  (s stride 2048)
  float* sT1 = sT0 + 32768;     // 4096  [s][a*32+q]
  float* sT2 = sT1 + 4096;      // 512   [s][r]
  float* sT3 = sT2 + 512;       // 4096  [s][n0*32+u]
  float* sT4 = sT0;             // 4096  [s*8+n0][v]  (aliases dead sT0)

  const int tid = threadIdx.x;
  const int wave = tid >> 5;
  const int bIdx = blockIdx.x >> 5;  // 32 tiles per batch (512/16)
  const int tile = blockIdx.x & 31;
  const int s0 = tile * 16;
  const int e = eidx[bIdx];

  const float* Xb = X + ((size_t)bIdx * 512 + s0) * 768;
  float* Ob = out + ((size_t)bIdx * 512 + s0) * 768;

  __builtin_prefetch(Xb, 0, 1);  // global_prefetch_b8

  // ---- stage cores into LDS (permuted so each stage's K dim is contiguous)
#if USE_ASYNC_LDS
  // Direct global->LDS DMA; per-lane LDS destinations allow permuted layouts.
  for (int i = tid; i < 384; i += 256)  // core0 [d][p], linear
    __builtin_amdgcn_global_load_async_to_lds_b32(
        (AS1 int*)(c0 + e * 384 + i), (AS3 int*)(sC0 + i), 0, 0);
  for (int i = tid; i < 8192; i += 256) {  // core1 [p][c][q] -> [(c,p)][q]
    int p = i >> 8, c = (i >> 5) & 7, q = i & 31;
    __builtin_amdgcn_global_load_async_to_lds_b32(
        (AS1 int*)(c1 + e * 8192 + i),
        (AS3 int*)(sC1 + (((c << 5) + p) << 5) + q), 0, 0);
  }
  for (int i = tid; i < 8192; i += 256) {  // core2 [q][a][r] -> [(a,q)][r]
    int q = i >> 8, a = (i >> 5) & 7, r = i & 31;
    __builtin_amdgcn_global_load_async_to_lds_b32(
        (AS1 int*)(c2 + e * 8192 + i),
        (AS3 int*)(sC2 + (((a << 5) + q) << 5) + r), 0, 0);
  }
  for (int i = tid * 4; i < 8192; i += 1024)  // core3 linear, 16B granules
    __builtin_amdgcn_global_load_async_to_lds_b128(
        (AS1 v4i*)(c3 + e * 8192 + i), (AS3 v4i*)(sC3 + i), 0, 0);
  for (int i = tid * 4; i < 8192; i += 1024)  // core4 linear, 16B granules
    __builtin_amdgcn_global_load_async_to_lds_b128(
        (AS1 v4i*)(c4 + e * 8192 + i), (AS3 v4i*)(sC4 + i), 0, 0);
  for (int i = tid; i < 512; i += 256) {  // core5 [v][12] -> [v][16] pad
    int v = i >> 4, n2 = i & 15;
    if (n2 < 12)
      __builtin_amdgcn_global_load_async_to_lds_b32(
          (AS1 int*)(c5 + e * 384 + v * 12 + n2),
          (AS3 int*)(sC5 + i), 0, 0);
    else
      sC5[i] = 0.0f;
  }
  __builtin_amdgcn_s_wait_asynccnt(0);
#else
  for (int i = tid; i < 384; i += 256) sC0[i] = c0[e * 384 + i];
  for (int i = tid; i < 8192; i += 256) {
    int p = i >> 8, c = (i >> 5) & 7, q = i & 31;
    sC1[(((c << 5) + p) << 5) + q] = c1[e * 8192 + i];
  }
  for (int i = tid; i < 8192; i += 256) {
    int q = i >> 8, a = (i >> 5) & 7, r = i & 31;
    sC2[(((a << 5) + q) << 5) + r] = c2[e * 8192 + i];
  }
  for (int i = tid; i < 8192; i += 256) sC3[i] = c3[e * 8192 + i];
  for (int i = tid; i < 8192; i += 256) sC4[i] = c4[e * 8192 + i];
  for (int i = tid; i < 512; i += 256) {
    int v = i >> 4, n2 = i & 15;
    sC5[i] = (n2 < 12) ? c5[e * 384 + v * 12 + n2] : 0.0f;
  }
#endif
  __syncthreads();

  // ---- stage 0: t0[(s,a,c)][p] = X[(s,a,c)][d] * C0[d][p]
  // X rows (s,a,c) are contiguous with stride 12 because 768 = 64*12.
  // M=1024, K=12, N=32 -> 64 mtiles x 2 ntiles.
  for (int job = wave; job < 128; job += NWAVE) {
    const int mt = job >> 1, nt = job & 1;
    v8f acc = zero8();
#pragma unroll
    for (int kk = 0; kk < 3; ++kk) {
      int lane = tid & 31;
      int m = lane & 15, kh = (lane >> 4) << 1;
      const float* ap = Xb + (mt * 16 + m) * 12 + kk * 4 + kh;
      v2f a; a.x = ap[0]; a.y = ap[1];
      v2f bf = load_b(sC0 + kk * 4 * 32 + nt * 16, 32);
      acc = wmma4(a, bf, acc);
    }
    store_c(sT0 + (mt * 16) * 32 + nt * 16, 32, acc);
  }
  __syncthreads();

  // ---- stage 1: t1[s][(a,q)] = sum_{c,p} t0[s][a][(c,p)] * C1[(c,p)][q]
  // per a: M=16, K=256, N=32 -> 8a x 2nt = 16 jobs.
  for (int job = wave; job < 16; job += NWAVE) {
    const int a = job >> 1, nt = job & 1;
    v8f acc = zero8();
    for (int kk = 0; kk < 64; ++kk) {
      v2f af = load_a(sT0 + a * 256 + kk * 4, 2048);
      v2f bf = load_b(sC1 + kk * 4 * 32 + nt * 16, 32);
      acc = wmma4(af, bf, acc);
    }
    store_c(sT1 + a * 32 + nt * 16, 256, acc);
  }
  __syncthreads();

  // ---- stage 2: t2[s][r] = sum_{a,q} t1[s][(a,q)] * C2[(a,q)][r]
  // M=16, K=256, N=32 -> 2 jobs.
  for (int job = wave; job < 2; job += NWAVE) {
    const int nt = job;
    v8f acc = zero8();
    for (int kk = 0; kk < 64; ++kk) {
      v2f af = load_a(sT1 + kk * 4, 256);
      v2f bf = load_b(sC2 + kk * 4 * 32 + nt * 16, 32);
      acc = wmma4(af, bf, acc);
    }
    store_c(sT2 + nt * 16, 32, acc);
  }
  __syncthreads();

  // ---- stage 3: t3[s][(n0,u)] = t2[s][r] * C3[r][(n0,u)]
  // M=16, K=32, N=256 -> 16 jobs.
  for (int job = wave; job < 16; job += NWAVE) {
    const int nt = job;
    v8f acc = zero8();
#pragma unroll
    for (int kk = 0; kk < 8; ++kk) {
      v2f af = load_a(sT2 + kk * 4, 32);
      v2f bf = load_b(sC3 + kk * 4 * 256 + nt * 16, 256);
      acc = wmma4(af, bf, acc);
    }
    store_c(sT3 + nt * 16, 256, acc);
  }
  __syncthreads();

  // ---- stages 4+5, fused per n1 to keep t4 at 16 KB:
  //   t4[(s,n0)][v] = t3[(s,n0)][u] * C4[u][(n1,v)]      (M=128,K=32,N=32)
  //   out[s][n0*96+n1*12+n2] = alpha * t4[(s,n0)][v] * C5[v][n2]  (N=12)
  for (int n1 = 0; n1 < 8; ++n1) {
    for (int job = wave; job < 16; job += NWAVE) {
      const int mt = job >> 1, nt = job & 1;
      v8f acc = zero8();
#pragma unroll
      for (int kk = 0; kk < 8; ++kk) {
        v2f af = load_a(sT3 + mt * 16 * 32 + kk * 4, 32);
        v2f bf = load_b(sC4 + kk * 4 * 256 + n1 * 32 + nt * 16, 256);
        acc = wmma4(af, bf, acc);
      }
      store_c(sT4 + mt * 16 * 32 + nt * 16, 32, acc);
    }
    __syncthreads();

    for (int job = wave; job < 8; job += NWAVE) {
      const int mt = job;
      v8f acc = zero8();
#pragma unroll
      for (int kk = 0; kk < 8; ++kk) {
        v2f af = load_a(sT4 + mt * 16 * 32 + kk * 4, 32);
        v2f bf = load_b(sC5 + kk * 4 * 16, 16);
        acc = wmma4(af, bf, acc);
      }
      // scatter: row = s*8 + n0 ; col = n2 (<12); scale by alpha=16
      int lane = tid & 31;
      int n2 = lane & 15, mb = (lane >> 4) << 3;
      if (n2 < 12) {
#pragma unroll
        for (int i = 0; i < 8; ++i) {
          int row = mt * 16 + mb + i;
          int s = row >> 3, n0 = row & 7;
          Ob[(size_t)s * 768 + n0 * 96 + n1 * 12 + n2] = acc[i] * 16.0f;
        }
      }
    }
    __syncthreads();
  }
}

// ---------------------------------------------------------------------------
extern "C" void kernel_launch(void* const* d_in, const int* in_sizes, int n_in,
                              void* d_out, int out_size, void* d_ws, size_t ws_size,
                              hipStream_t stream) {
  const float* X  = (const float*)d_in[0];
  const float* rw = (const float*)d_in[1];
  const float* rb = (const float*)d_in[2];
  const float* un = (const float*)d_in[3];
  const float* c0 = (const float*)d_in[4];
  const float* c1 = (const float*)d_in[5];
  const float* c2 = (const float*)d_in[6];
  const float* c3 = (const float*)d_in[7];
  const float* c4 = (const float*)d_in[8];
  const float* c5 = (const float*)d_in[9];
  float* out = (float*)d_out;
  int* eidx = (int*)d_ws;  // 32 ints

  router_kernel<<<32, 256, 0, stream>>>(X, rw, rb, un, eidx);

  const size_t shmem = (size_t)75136 * sizeof(float);  // 293.5 KB / WGP
  tt_moe_kernel<<<32 * 32, 256, shmem, stream>>>(X, c0, c1, c2, c3, c4, c5,
                                                 eidx, out);
  (void)in_sizes; (void)n_in; (void)out_size; (void)ws_size;
}